// roi_pooling_40080634806597
// MI455X (gfx1250) — compile-verified
//
#include <hip/hip_runtime.h>

// ROI pooling: crop_and_resize (bilinear, 14x14) + 2x2 maxpool -> (N,7,7,512)
// Feature map (1,50,75,512) fp32 NHWC; rois (N,4) = [x1,y1,x2,y2] in image px.
//
// MI455X design (compile-only reasoning):
//  - Memory floor: 100MB streamed fp32 output ~4.3us @ 23.3TB/s; gathers hit
//    the L2/WGP$-resident 7.68MB feature map. No shared-K matmul structure ->
//    VALU FMA + coalesced b128 vmem, not v_wmma (would only add issue).
//  - 64-thread blocks (2 waves, wave32), 8 channels/thread: per-wave scalar
//    work (ROI divides, sample setup, gather bases) covers 2x channels; each
//    corner pixel = two global_load_b128 off one scalar base (+16B imm).
//  - OOB validity mask folded into the block-uniform combined weights
//    (weights >= 0, inputs finite -> zero-weighted sum == +0.0 exactly,
//    matching the reference's where(valid, v, 0)). No per-channel masking.
//  - 2x2 maxpool as a 3-op max tree over the 4 candidates (no -inf init).
//  - One block per pooled output row (n, py); px loop fully unrolled for a
//    deep (224-load) pipeline against L2 latency.
//  - Output stored non-temporally (TH=NT): the write-once 100MB stream must
//    not evict the hot feature map from the 192MB L2.

#define POOLSZ 7
#define CROPSZ 14
#define FMAP_H 50
#define FMAP_W 75
#define FMAP_C 512

typedef float v4f __attribute__((ext_vector_type(4)));

static __device__ __forceinline__ v4f vmax4(v4f a, v4f b) {
    v4f r;
    r.x = fmaxf(a.x, b.x);
    r.y = fmaxf(a.y, b.y);
    r.z = fmaxf(a.z, b.z);
    r.w = fmaxf(a.w, b.w);
    return r;
}

__global__ __launch_bounds__(64) void roi_pool_bilinear_max_kernel(
    const float* __restrict__ fmap,      // (1,H,W,C)
    const float* __restrict__ rois,      // (N,4) x1,y1,x2,y2
    const int*   __restrict__ img_size,  // [img_h, img_w]
    float* __restrict__ out,             // (N,7,7,512)
    int N)
{
    const int bid = blockIdx.x;
    const int py  = bid % POOLSZ;
    const int n   = bid / POOLSZ;
    if (n >= N) return;

    // ---- block-uniform ROI math (scalarized: s_load + scalar ops) ----
    const float img_h = (float)(img_size[0] - 1);   // 799
    const float img_w = (float)(img_size[1] - 1);   // 1199
    const float rx1 = rois[n * 4 + 0];
    const float ry1 = rois[n * 4 + 1];
    const float rx2 = rois[n * 4 + 2];
    const float ry2 = rois[n * 4 + 3];

    const float fh = (float)(FMAP_H - 1);           // 49
    const float fw = (float)(FMAP_W - 1);           // 74

    // normalized box [y1,x1,y2,x2] exactly as reference
    const float by1 = ry1 / img_h, bx1 = rx1 / img_w;
    const float by2 = ry2 / img_h, bx2 = rx2 / img_w;

    const float ybase = by1 * fh;
    const float ystep = (by2 - by1) * fh / (float)(CROPSZ - 1);
    const float xbase = bx1 * fw;
    const float xstep = (bx2 - bx1) * fw / (float)(CROPSZ - 1);

    // ---- hoist both y-samples for this pooled row (block-uniform) ----
    float wy_a[2];      // bilinear y-weight per crop row
    float ym_a[2];      // validity mask per crop row (folded into weights)
    int   r0_a[2];      // row offset of y0, pre-scaled to elements (row*W*C)
    int   r1_a[2];      // row offset of y1, pre-scaled to elements
    #pragma unroll
    for (int dy = 0; dy < 2; ++dy) {
        const float yy  = ybase + (float)(2 * py + dy) * ystep;
        ym_a[dy] = (yy >= 0.0f && yy <= fh) ? 1.0f : 0.0f;
        const float y0f = floorf(yy);
        wy_a[dy] = yy - y0f;
        int y0i = (int)y0f;
        y0i = min(max(y0i, 0), FMAP_H - 1);
        r0_a[dy] = y0i * (FMAP_W * FMAP_C);
        r1_a[dy] = min(y0i + 1, FMAP_H - 1) * (FMAP_W * FMAP_C);
    }

    const int c0 = threadIdx.x * 8;                 // 8 channels per lane
    const float* base = fmap + c0;
    float* outp = out + ((size_t)(n * (POOLSZ * POOLSZ) + py * POOLSZ) * FMAP_C + c0);

    #pragma unroll
    for (int px = 0; px < POOLSZ; ++px) {
        v4f vA[2][2];   // candidate values, low 4 channels
        v4f vB[2][2];   // candidate values, high 4 channels

        #pragma unroll
        for (int dx = 0; dx < 2; ++dx) {
            const float xx  = xbase + (float)(2 * px + dx) * xstep;
            const float xm  = (xx >= 0.0f && xx <= fw) ? 1.0f : 0.0f;
            const float x0f = floorf(xx);
            const float wx  = xx - x0f;
            const float omwx = 1.0f - wx;
            int x0i = (int)x0f;
            x0i = min(max(x0i, 0), FMAP_W - 1);
            const int x1i = min(x0i + 1, FMAP_W - 1);
            const int xe0 = x0i * FMAP_C;           // col offset pre-scaled
            const int xe1 = x1i * FMAP_C;

            #pragma unroll
            for (int dy = 0; dy < 2; ++dy) {
                // 4 unique pixels; each = two b128 loads off one scalar base
                const float* ptl  = base + (size_t)(r0_a[dy] + xe0);
                const float* ptr_ = base + (size_t)(r0_a[dy] + xe1);
                const float* pbl  = base + (size_t)(r1_a[dy] + xe0);
                const float* pbr  = base + (size_t)(r1_a[dy] + xe1);

                const v4f tlA = *(const v4f*)(ptl);
                const v4f tlB = *(const v4f*)(ptl + 4);
                const v4f trA = *(const v4f*)(ptr_);
                const v4f trB = *(const v4f*)(ptr_ + 4);
                const v4f blA = *(const v4f*)(pbl);
                const v4f blB = *(const v4f*)(pbl + 4);
                const v4f brA = *(const v4f*)(pbr);
                const v4f brB = *(const v4f*)(pbr + 4);

                // combined corner weights with validity mask folded in
                // (all block-uniform scalar math)
                const float wy   = wy_a[dy];
                const float omwy = 1.0f - wy;
                const float m    = xm * ym_a[dy];
                const float wtl = omwy * omwx * m;
                const float wtr = omwy * wx   * m;
                const float wbl = wy * omwx   * m;
                const float wbr = wy * wx     * m;

                // 4 FMAs per channel, no per-channel masking
                v4f valA = wbr * brA;
                valA = wbl * blA + valA;
                valA = wtr * trA + valA;
                valA = wtl * tlA + valA;

                v4f valB = wbr * brB;
                valB = wbl * blB + valB;
                valB = wtr * trB + valB;
                valB = wtl * tlB + valB;

                vA[dx][dy] = valA;
                vB[dx][dy] = valB;
            }
        }

        // 2x2 maxpool: 3-op max tree
        const v4f accA = vmax4(vmax4(vA[0][0], vA[0][1]), vmax4(vA[1][0], vA[1][1]));
        const v4f accB = vmax4(vmax4(vB[0][0], vB[0][1]), vmax4(vB[1][0], vB[1][1]));

        // streamed, write-once output: non-temporal b128 stores
        float* dst = outp + (size_t)px * FMAP_C;
        __builtin_nontemporal_store(accA, (v4f*)(dst));
        __builtin_nontemporal_store(accB, (v4f*)(dst + 4));
    }
}

extern "C" void kernel_launch(void* const* d_in, const int* in_sizes, int n_in,
                              void* d_out, int out_size, void* d_ws, size_t ws_size,
                              hipStream_t stream) {
    const float* fmap     = (const float*)d_in[0];   // (1,50,75,512) fp32
    const float* rois     = (const float*)d_in[1];   // (N,4) fp32
    const int*   img_size = (const int*)d_in[2];     // [800,1200] int32
    float* out = (float*)d_out;                      // (N,7,7,512) fp32

    const int N = in_sizes[1] / 4;
    const int blocks = N * POOLSZ;                   // one block per pooled output row
    roi_pool_bilinear_max_kernel<<<blocks, 64, 0, stream>>>(fmap, rois, img_size, out, N);
}